// ModalityAttentionLayer_82635170775340
// MI455X (gfx1250) — compile-verified
//
#include <hip/hip_runtime.h>
#include <hip/hip_bf16.h>
#include <math.h>

typedef __attribute__((ext_vector_type(2))) float v2f;
typedef __attribute__((ext_vector_type(8))) float v8f;

#define B_      16
#define C_      4
#define HW_     (512 * 512)
#define PLANES  (B_ * C_)   // 64 (b,c) planes
#define CHUNKS  16          // partial-sum chunks per plane
#define THREADS 256

// ---------------------------------------------------------------------------
// Kernel 1: partial mean-pool. Grid = PLANES*CHUNKS blocks; each block reduces
// HW/CHUNKS = 16384 floats (4096 float4, 16 per thread) of one plane and
// writes one partial to d_ws. Fixed-order tree reduction => deterministic.
// ---------------------------------------------------------------------------
__global__ __launch_bounds__(THREADS)
void pool_partial_kernel(const float* __restrict__ x, float* __restrict__ partials) {
  const int plane = blockIdx.x >> 4;   // 0..63
  const int chunk = blockIdx.x & 15;   // 0..15
  const float4* xp = reinterpret_cast<const float4*>(x)
                     + (size_t)plane * (HW_ / 4) + (size_t)chunk * 4096;
  float s = 0.0f;
#pragma unroll
  for (int i = 0; i < 16; ++i) {
    float4 v = xp[threadIdx.x + i * THREADS];   // global_load_b128, coalesced
    s += (v.x + v.y) + (v.z + v.w);
  }
  __shared__ float sdata[THREADS];
  sdata[threadIdx.x] = s;
  __syncthreads();
  for (int off = THREADS / 2; off > 0; off >>= 1) {
    if (threadIdx.x < (unsigned)off) sdata[threadIdx.x] += sdata[threadIdx.x + off];
    __syncthreads();
  }
  if (threadIdx.x == 0) partials[blockIdx.x] = sdata[0];  // index = plane*16+chunk
}

// ---------------------------------------------------------------------------
// Kernel 2: single wave32. Finishes the mean, then the 2-layer MLP entirely
// with V_WMMA_F32_16X16X4_F32.
//   GEMM1: h(16x16) = pooled(16x4) x w1^T(4x16)         -> 1 WMMA
//   GEMM2: gate_pre(16x4pad16) = h(16x16) x w2^T(16x16) -> 4 chained WMMAs
// Operand layouts per CDNA5 ISA 7.12.2:
//   A f32 16x4 : lane L, vgpr j  -> A[L%16][2*(L/16)+j]
//   B f32 4x16 : lane L, vgpr j  -> B[2*(L/16)+j][L%16]
//   C/D 16x16  : lane L, vgpr v  -> D[v+8*(L/16)][L%16]
// ---------------------------------------------------------------------------
__global__ __launch_bounds__(32)
void mlp_wmma_kernel(const float* __restrict__ partials,
                     const float* __restrict__ w1, const float* __restrict__ b1,
                     const float* __restrict__ w2, const float* __restrict__ b2,
                     float* __restrict__ weights_out) {
  const int lane = threadIdx.x;
  const int half = lane >> 4;   // 0 | 1
  const int lm   = lane & 15;
  const float inv_hw = 1.0f / (float)HW_;

  // A = pooled: M = lm (batch), K = 2*half + j (channel)
  v2f a;
#pragma unroll
  for (int j = 0; j < 2; ++j) {
    const int K = 2 * half + j;
    const float* p = partials + (lm * C_ + K) * CHUNKS;
    float s = 0.0f;
#pragma unroll
    for (int t = 0; t < CHUNKS; ++t) s += p[t];
    a[j] = s * inv_hw;
  }
  // B = w1^T: N = lm (o), K = 2*half + j (c); w1 is (16,4) row-major [o][c]
  v2f bm;
#pragma unroll
  for (int j = 0; j < 2; ++j) bm[j] = w1[lm * C_ + (2 * half + j)];

  v8f acc = {};
  v8f h = __builtin_amdgcn_wmma_f32_16x16x4_f32(false, a, false, bm,
                                                (short)0, acc, false, false);

  // bias + relu, stage h row-major in LDS to re-layout D -> A for GEMM2
  __shared__ float hmat[16 * 16];
  const float bias1 = b1[lm];                      // N = lm
#pragma unroll
  for (int v = 0; v < 8; ++v) {
    const int M = v + 8 * half;
    float hv = h[v] + bias1;
    hmat[M * 16 + lm] = hv > 0.0f ? hv : 0.0f;
  }
  __syncthreads();

  // GEMM2: K=16 as 4 chained 16x16x4 WMMAs; B padded with zeros for N>=4
  v8f d2 = {};
#pragma unroll
  for (int kk = 0; kk < 4; ++kk) {
    v2f a2, b2v;
#pragma unroll
    for (int j = 0; j < 2; ++j) {
      const int K = 4 * kk + 2 * half + j;                   // o index
      a2[j]  = hmat[lm * 16 + K];                            // M = lm (batch)
      b2v[j] = (lm < C_) ? w2[lm * 16 + K] : 0.0f;           // N = lm (c); w2[c][o]
    }
    d2 = __builtin_amdgcn_wmma_f32_16x16x4_f32(false, a2, false, b2v,
                                               (short)0, d2, false, false);
  }

  // sigmoid + residual floor; D layout: M = v+8*half (batch), N = lm (channel)
  if (lm < C_) {
    const float bias2 = b2[lm];
#pragma unroll
    for (int v = 0; v < 8; ++v) {
      const int M = v + 8 * half;
      float g = 1.0f / (1.0f + expf(-(d2[v] + bias2)));
      weights_out[M * C_ + lm] = 0.1f + 0.9f * g;
    }
  }
}

// ---------------------------------------------------------------------------
// Kernel 3: out = x * weights[plane]. Pure streaming pass; x should hit L2
// (64 MiB x fits in the 192 MB device L2 after the pooling pass).
// Grid = PLANES*64 blocks, 256 threads, 4 float4 each (1024 float4 / block).
// ---------------------------------------------------------------------------
__global__ __launch_bounds__(THREADS)
void scale_kernel(const float* __restrict__ x, const float* __restrict__ weights,
                  float* __restrict__ out) {
  const int plane = blockIdx.x >> 6;   // 0..63
  const int chunk = blockIdx.x & 63;   // 0..63
  const float wt = weights[plane];     // wave-uniform scalar load
  const size_t base = (size_t)plane * (HW_ / 4) + (size_t)chunk * 1024;
  const float4* xp = reinterpret_cast<const float4*>(x) + base;
  float4* op = reinterpret_cast<float4*>(out) + base;
#pragma unroll
  for (int i = 0; i < 4; ++i) {
    const int idx = threadIdx.x + i * THREADS;
    float4 v = xp[idx];
    float4 r;
    r.x = v.x * wt; r.y = v.y * wt; r.z = v.z * wt; r.w = v.w * wt;
    op[idx] = r;                      // global_store_b128
  }
}

// ---------------------------------------------------------------------------
extern "C" void kernel_launch(void* const* d_in, const int* in_sizes, int n_in,
                              void* d_out, int out_size, void* d_ws, size_t ws_size,
                              hipStream_t stream) {
  const float* x  = (const float*)d_in[0];   // (16,4,512,512)
  const float* w1 = (const float*)d_in[1];   // (16,4,1,1)
  const float* b1 = (const float*)d_in[2];   // (16,)
  const float* w2 = (const float*)d_in[3];   // (4,16,1,1)
  const float* b2 = (const float*)d_in[4];   // (4,)
  // d_in[5..7] (embed_w, affine_a, affine_b) feed only the noise path, which
  // is multiplied by 0.0 in the reference -> dead.

  float* out         = (float*)d_out;
  float* weights_out = out + (size_t)PLANES * HW_;  // 2nd tuple output (64 floats)
  float* partials    = (float*)d_ws;                // 64*16 floats = 4 KiB scratch

  pool_partial_kernel<<<dim3(PLANES * CHUNKS), dim3(THREADS), 0, stream>>>(x, partials);
  mlp_wmma_kernel<<<dim3(1), dim3(32), 0, stream>>>(partials, w1, b1, w2, b2, weights_out);
  scale_kernel<<<dim3(PLANES * 64), dim3(THREADS), 0, stream>>>(x, weights_out, out);
}